// SymbolicPlanner_80822694576135
// MI455X (gfx1250) — compile-verified
//
#include <hip/hip_runtime.h>
#include <hip/hip_bf16.h>
#include <math.h>

// ---------------------------------------------------------------------------
// Problem constants (from reference)
// ---------------------------------------------------------------------------
#define GRID_N   362
#define B_ROWS   4096
#define A_TABLES 3
#define N_KEYS   10000
#define KDIM     6
#define PDIM     7
#define EPSF     1e-8f

#define N_TILES  (N_KEYS / 16)   // 625 exactly
#define WAVES    4               // waves per 16-row group (wave32)

typedef __attribute__((ext_vector_type(2))) float v2f;
typedef __attribute__((ext_vector_type(8))) float v8f;

// ---------------------------------------------------------------------------
// Workspace layout (byte offsets, 16-aligned). Total ~1.03 MB.
// ---------------------------------------------------------------------------
#define KNT_OFF   0u                         // float[3*6*10000]  normalized keys, k-major
#define QN_OFF    720000u                    // float[4096*6]     normalized queries
#define CI_OFF    818304u                    // int  [4096*6]     cell indices
#define RL_OFF    916608u                    // int  [4160]       bucketed row list
#define META_OFF  933248u                    // int  [8]          off0..off3, c0..c2
#define DL_OFF    933280u                    // float[4096*6]     deltas (pred[:, :6])

// ---------------------------------------------------------------------------
// K1: normalize keys, store transposed knT[a][k][n] for coalesced B-tile loads
// ---------------------------------------------------------------------------
__global__ void k_norm_keys(const float* __restrict__ keys, float* __restrict__ knT) {
    int r = blockIdx.x * blockDim.x + threadIdx.x;        // 0 .. 3*10000-1
    if (r >= A_TABLES * N_KEYS) return;
    int a = r / N_KEYS;
    int n = r - a * N_KEYS;
    const float* kr = keys + (size_t)r * KDIM;
    float k0 = kr[0], k1 = kr[1], k2 = kr[2], k3 = kr[3], k4 = kr[4], k5 = kr[5];
    float nrm = sqrtf(k0*k0 + k1*k1 + k2*k2 + k3*k3 + k4*k4 + k5*k5) + EPSF;
    float inv = 1.0f / nrm;
    float* base = knT + (size_t)a * KDIM * N_KEYS + n;
    base[0 * N_KEYS] = k0 * inv;
    base[1 * N_KEYS] = k1 * inv;
    base[2 * N_KEYS] = k2 * inv;
    base[3 * N_KEYS] = k3 * inv;
    base[4 * N_KEYS] = k4 * inv;
    base[5 * N_KEYS] = k5 * inv;
}

// ---------------------------------------------------------------------------
// K2: per-row argmax (first max wins, like jnp.argmax), cell indices, qn
// ---------------------------------------------------------------------------
__device__ __forceinline__ int bindc(int i) {
    return i < 1 ? 1 : (i > GRID_N - 1 ? GRID_N - 1 : i);
}

__global__ void k_row_prep(const float* __restrict__ x,
                           float* __restrict__ qn, int* __restrict__ cellIdx) {
    int b = blockIdx.x * blockDim.x + threadIdx.x;
    if (b >= B_ROWS) return;
    const float* xr = x + (size_t)b * GRID_N;
    int   ptr  = 0;
    float best = xr[0];
    for (int i = 1; i < GRID_N; ++i) {
        float v = xr[i];
        if (v > best) { best = v; ptr = i; }   // strict > keeps first occurrence
    }
    int ci[6];
    ci[0] = 0;
    ci[1] = ptr;
    ci[2] = bindc(ptr - 19);
    ci[3] = bindc(ptr + 19);
    ci[4] = bindc(ptr - 1);
    ci[5] = bindc(ptr + 1);
    float at[6], ss = 0.0f;
    #pragma unroll
    for (int j = 0; j < 6; ++j) { at[j] = xr[ci[j]]; ss += at[j] * at[j]; }
    float inv = 1.0f / (sqrtf(ss) + EPSF);
    #pragma unroll
    for (int j = 0; j < 6; ++j) {
        qn[b * 6 + j]      = at[j] * inv;
        cellIdx[b * 6 + j] = ci[j];
    }
}

// ---------------------------------------------------------------------------
// K3: deterministic bucketing by batch_a (rank scan, no atomics).
// rowlist[off[a] + rank] = b ; offsets padded to multiples of 16.
// ---------------------------------------------------------------------------
__global__ void k_bucket(const int* __restrict__ ba,
                         int* __restrict__ rowlist, int* __restrict__ meta) {
    int b = blockIdx.x * blockDim.x + threadIdx.x;
    if (b >= B_ROWS) return;
    int mya = ba[b];
    int rank = 0, c0 = 0, c1 = 0, c2 = 0;
    for (int i = 0; i < B_ROWS; ++i) {
        int ai = ba[i];
        c0 += (ai == 0);
        c1 += (ai == 1);
        c2 += (ai == 2);
        rank += (ai == mya) && (i < b);
    }
    int off1 = ((c0 + 15) >> 4) << 4;
    int off2 = off1 + (((c1 + 15) >> 4) << 4);
    int off3 = off2 + (((c2 + 15) >> 4) << 4);
    int offa = (mya == 0) ? 0 : (mya == 1) ? off1 : off2;
    rowlist[offa + rank] = b;
    if (b == 0) {
        meta[0] = 0;  meta[1] = off1; meta[2] = off2; meta[3] = off3;
        meta[4] = c0; meta[5] = c1;   meta[6] = c2;   meta[7] = 0;
    }
}

// ---------------------------------------------------------------------------
// K4: main attention. One block = one 16-row group (all rows share table a).
// sim tile (16x16, f32) via two chained V_WMMA_F32_16X16X4_F32 (K=6, zero-pad
// k=6,7). sim in [-1,1] -> softmax without max-subtraction: accumulate
// sum(e), sum(e*v[0..6]), max(sim) in a single pass. 4 waves split 625 tiles.
// ---------------------------------------------------------------------------
__global__ void __launch_bounds__(WAVES * 32)
k_attn(const float* __restrict__ knT, const float* __restrict__ values,
       const float* __restrict__ qn,  const int* __restrict__ rowlist,
       const int* __restrict__ meta,  float* __restrict__ deltas,
       float* __restrict__ out_unc,   float* __restrict__ out_done) {
    __shared__ float part[WAVES][16][9];   // per-wave partials: acc[7], sumexp, max

    const int off1 = meta[1], off2 = meta[2], off3 = meta[3];
    const int s0 = blockIdx.x * 16;
    if (s0 >= off3) return;                                  // uniform early exit
    const int a    = (s0 >= off2) ? 2 : (s0 >= off1) ? 1 : 0;
    const int offa = (a == 0) ? 0 : (a == 1) ? off1 : off2;
    const int cnta = meta[4 + a];

    const int lane = threadIdx.x & 31;
    const int wave = threadIdx.x >> 5;
    const int m    = lane & 15;       // A: row M=m ; B/C: col N=m
    const int hi   = lane >> 4;       // 0: K=0/1 (A,B), rows 0-7 (C); 1: K=2/3, rows 8-15

    // A-matrix (16x4 f32): VGPR0 = K=(0|2), VGPR1 = K=(1|3); chunk2: K=(4|6),(5|7), 6/7 = 0
    const int  lrank = s0 - offa + m;
    const bool valid = lrank < cnta;
    const int  brow  = valid ? rowlist[s0 + m] : 0;
    const float vm   = valid ? 1.0f : 0.0f;                  // padding rows -> qn=0
    const float* q   = qn + brow * 6;
    v2f A1, A2;
    A1.x = q[hi ? 2 : 0] * vm;
    A1.y = q[hi ? 3 : 1] * vm;
    A2.x = hi ? 0.0f : q[4] * vm;
    A2.y = hi ? 0.0f : q[5] * vm;

    const float* knTa = knT + (size_t)a * KDIM * N_KEYS;
    const float* va   = values + (size_t)a * N_KEYS * PDIM;

    float acc[8][PDIM];
    float sume[8], smax[8];
    #pragma unroll
    for (int r = 0; r < 8; ++r) {
        sume[r] = 0.0f; smax[r] = -1.0e30f;
        #pragma unroll
        for (int p = 0; p < PDIM; ++p) acc[r][p] = 0.0f;
    }

    for (int t = wave; t < N_TILES; t += WAVES) {
        const int n = t * 16 + m;
        // B-matrix (4x16 f32): mirrors A layout across lane halves
        v2f B1, B2;
        B1.x = knTa[(hi ? 2 : 0) * N_KEYS + n];
        B1.y = knTa[(hi ? 3 : 1) * N_KEYS + n];
        B2.x = hi ? 0.0f : knTa[4 * N_KEYS + n];
        B2.y = hi ? 0.0f : knTa[5 * N_KEYS + n];

        const float* vr = va + (size_t)n * PDIM;             // this lane's value row
        float v0 = vr[0], v1 = vr[1], v2 = vr[2], v3 = vr[3];
        float v4 = vr[4], v5 = vr[5], v6 = vr[6];

        v8f c = {0.f, 0.f, 0.f, 0.f, 0.f, 0.f, 0.f, 0.f};
        c = __builtin_amdgcn_wmma_f32_16x16x4_f32(false, A1, false, B1,
                                                  (short)0, c, false, false);
        c = __builtin_amdgcn_wmma_f32_16x16x4_f32(false, A2, false, B2,
                                                  (short)0, c, false, false);

        // C layout: element r = sim[row r+8*hi][col m]; accumulate softmax terms
        #pragma unroll
        for (int r = 0; r < 8; ++r) {
            float s = c[r];
            float e = __expf(s);                             // s in [-1,1]: no overflow
            sume[r] += e;
            smax[r] = fmaxf(smax[r], s);
            acc[r][0] += e * v0; acc[r][1] += e * v1; acc[r][2] += e * v2;
            acc[r][3] += e * v3; acc[r][4] += e * v4; acc[r][5] += e * v5;
            acc[r][6] += e * v6;
        }
    }

    // Butterfly reduce across the 16 lanes of each half (xor 8,4,2,1 stays in-half)
    #pragma unroll
    for (int d = 8; d >= 1; d >>= 1) {
        #pragma unroll
        for (int r = 0; r < 8; ++r) {
            sume[r] += __shfl_xor(sume[r], d, 32);
            smax[r] = fmaxf(smax[r], __shfl_xor(smax[r], d, 32));
            #pragma unroll
            for (int p = 0; p < PDIM; ++p) acc[r][p] += __shfl_xor(acc[r][p], d, 32);
        }
    }

    if (m == 0) {                                            // lanes 0 and 16 write
        #pragma unroll
        for (int r = 0; r < 8; ++r) {
            const int row = r + 8 * hi;
            #pragma unroll
            for (int p = 0; p < PDIM; ++p) part[wave][row][p] = acc[r][p];
            part[wave][row][7] = sume[r];
            part[wave][row][8] = smax[r];
        }
    }
    __syncthreads();

    if (threadIdx.x < 16) {
        const int mm = threadIdx.x;
        const int lr = s0 - offa + mm;
        if (lr < cnta) {
            const int b = rowlist[s0 + mm];
            float se = 0.0f, mx = -1.0e30f, ac[PDIM];
            #pragma unroll
            for (int p = 0; p < PDIM; ++p) ac[p] = 0.0f;
            #pragma unroll
            for (int w = 0; w < WAVES; ++w) {
                se += part[w][mm][7];
                mx = fmaxf(mx, part[w][mm][8]);
                #pragma unroll
                for (int p = 0; p < PDIM; ++p) ac[p] += part[w][mm][p];
            }
            const float inv = 1.0f / se;
            #pragma unroll
            for (int p = 0; p < 6; ++p) deltas[b * 6 + p] = ac[p] * inv;
            out_done[b] = ac[6] * inv;
            out_unc[b]  = -mx;
        }
    }
}

// ---------------------------------------------------------------------------
// K5: outputs = batch_x + scatter(deltas) ; .set semantics => highest j wins
// ---------------------------------------------------------------------------
__global__ void k_out(const float* __restrict__ x, const int* __restrict__ cellIdx,
                      const float* __restrict__ deltas, float* __restrict__ out) {
    const int b = blockIdx.x;
    __shared__ int   ci[6];
    __shared__ float dl[6];
    if (threadIdx.x < 6) {
        ci[threadIdx.x] = cellIdx[b * 6 + threadIdx.x];
        dl[threadIdx.x] = deltas[b * 6 + threadIdx.x];
    }
    __syncthreads();
    for (int i = threadIdx.x; i < GRID_N; i += blockDim.x) {
        float d = 0.0f;
        #pragma unroll
        for (int j = 5; j >= 0; --j) {
            if (ci[j] == i) { d = dl[j]; break; }            // last .set wins
        }
        out[(size_t)b * GRID_N + i] = x[(size_t)b * GRID_N + i] + d;
    }
}

// ---------------------------------------------------------------------------
// Host-side launcher
// ---------------------------------------------------------------------------
extern "C" void kernel_launch(void* const* d_in, const int* in_sizes, int n_in,
                              void* d_out, int out_size, void* d_ws, size_t ws_size,
                              hipStream_t stream) {
    const float* batch_x = (const float*)d_in[0];   // 4096 x 362
    const int*   batch_a = (const int*)  d_in[1];   // 4096
    const float* keys    = (const float*)d_in[2];   // 3 x 10000 x 6
    const float* values  = (const float*)d_in[3];   // 3 x 10000 x 7

    char* ws = (char*)d_ws;
    float* knT     = (float*)(ws + KNT_OFF);
    float* qn      = (float*)(ws + QN_OFF);
    int*   cellIdx = (int*)  (ws + CI_OFF);
    int*   rowlist = (int*)  (ws + RL_OFF);
    int*   meta    = (int*)  (ws + META_OFF);
    float* deltas  = (float*)(ws + DL_OFF);

    float* out      = (float*)d_out;                       // 4096*362 outputs
    float* out_unc  = out + (size_t)B_ROWS * GRID_N;       // 4096 uncertainties
    float* out_done = out_unc + B_ROWS;                    // 4096 dones

    // K1: normalize keys
    {
        int n = A_TABLES * N_KEYS;
        k_norm_keys<<<(n + 255) / 256, 256, 0, stream>>>(keys, knT);
    }
    // K2: argmax / cell indices / qn
    k_row_prep<<<(B_ROWS + 255) / 256, 256, 0, stream>>>(batch_x, qn, cellIdx);
    // K3: deterministic bucketing by a
    k_bucket<<<(B_ROWS + 255) / 256, 256, 0, stream>>>(batch_a, rowlist, meta);
    // K4: WMMA attention (max padded rows: 4096 + 3*16 => 259 groups of 16)
    k_attn<<<259, WAVES * 32, 0, stream>>>(knT, values, qn, rowlist, meta,
                                           deltas, out_unc, out_done);
    // K5: scatter-add into outputs
    k_out<<<B_ROWS, 384, 0, stream>>>(batch_x, cellIdx, deltas, out);
}